// EquivariantGraphConvCheap_58557584114044
// MI455X (gfx1250) — compile-verified
//
#include <hip/hip_runtime.h>

typedef __attribute__((ext_vector_type(2))) float v2f;
typedef __attribute__((ext_vector_type(8))) float v8f;

#define H   128
#define ROW 512   // 4 channels * 128 floats per node

// ---------------------------------------------------------------------------
// Kernel 1: zero d_out (it doubles as the segment-sum accumulator).
// ---------------------------------------------------------------------------
__global__ void zero_f4(float4* __restrict__ p, long long n4) {
  long long i = (long long)blockIdx.x * blockDim.x + threadIdx.x;
  long long stride = (long long)gridDim.x * blockDim.x;
  float4 z = make_float4(0.f, 0.f, 0.f, 0.f);
  for (; i < n4; i += stride) p[i] = z;
}

// ---------------------------------------------------------------------------
// Kernel 2: edge scatter-add of RAW features (linearity lets us defer W_rel).
// 128 threads per edge, each moves one float4 (512 floats = 4ch x 128).
// ---------------------------------------------------------------------------
__global__ void __launch_bounds__(256)
edge_scatter(const float* __restrict__ x,
             const long long* __restrict__ ei,
             float* __restrict__ agg, int E) {
  int e = blockIdx.x * 2 + (threadIdx.x >> 7);
  if (e >= E) return;
  int t = threadIdx.x & 127;
  long long dst_n = ei[e];                  // row (destination / segment id)
  long long src_n = ei[(long long)E + e];   // col (gather source)
  float4 v = ((const float4*)(x + src_n * ROW))[t];
  float* d = agg + dst_n * ROW + (long long)t * 4;
  atomicAdd(d + 0, v.x);
  atomicAdd(d + 1, v.y);
  atomicAdd(d + 2, v.z);
  atomicAdd(d + 3, v.w);
}

// ---------------------------------------------------------------------------
// Kernel 3: out = X @ Wroot^T + AGG @ Wrel^T (+bias for c==0), f32 WMMA.
// Block = 256 thr = 8 wave32, handles one (16-node tile, channel).
// Stage X and AGG tiles in LDS, barrier, then overwrite out rows in place.
// ---------------------------------------------------------------------------
__global__ void __launch_bounds__(256)
gemm_combine(const float* __restrict__ x,
             const float* __restrict__ Wsrel, const float* __restrict__ Wsroot,
             const float* __restrict__ bs,
             const float* __restrict__ Wvrel, const float* __restrict__ Wvroot,
             float* __restrict__ out, int N) {
  __shared__ float Xs[16 * H];
  __shared__ float Gs[16 * H];

  const int tile = blockIdx.x;
  const int c    = blockIdx.y;          // channel 0..3
  const int tid  = threadIdx.x;

  const float* Wroot = (c == 0) ? Wsroot : Wvroot;
  const float* Wrel  = (c == 0) ? Wsrel  : Wvrel;

  const bool fullTile = ((long long)tile * 16 + 16) <= (long long)N;

  // Stage 16 rows of X and AGG (agg currently lives in `out`) into LDS.
  for (int f = tid * 4; f < 16 * H; f += 256 * 4) {
    int r = f >> 7;                      // row within tile
    int h = f & (H - 1);
    long long node = (long long)tile * 16 + r;
    if (node >= N) node = N - 1;         // clamp (N%16==0 in practice)
    long long base = node * ROW + (long long)c * H + h;
    *(float4*)&Xs[f] = *(const float4*)(x   + base);
    *(float4*)&Gs[f] = *(const float4*)(out + base);
  }
  __syncthreads();

  const int lane    = tid & 31;
  const int wave    = tid >> 5;
  const int nc      = lane & 15;         // A-row m / B-col n for this lane
  const int kh      = lane >> 4;         // k-half select (16x16x4 f32 layout)
  const int colBase = wave * 16;         // this wave's 16 output columns

  const float* wr = Wroot + (long long)(colBase + nc) * H;  // B[k][n] = W[n][k]
  const float* wl = Wrel  + (long long)(colBase + nc) * H;

  v8f acc = {};
  #pragma unroll 4
  for (int kb = 0; kb < H / 4; ++kb) {
    int k = kb * 4 + kh * 2;
    v2f aX = *(const v2f*)&Xs[nc * H + k];   // A: X rows (LDS)
    v2f aG = *(const v2f*)&Gs[nc * H + k];   // A: AGG rows (LDS)
    v2f bR = *(const v2f*)&wr[k];            // B: Wroot row (global, L2-hot)
    v2f bL = *(const v2f*)&wl[k];            // B: Wrel  row
    acc = __builtin_amdgcn_wmma_f32_16x16x4_f32(false, aX, false, bR,
                                                (short)0, acc, false, false);
    acc = __builtin_amdgcn_wmma_f32_16x16x4_f32(false, aG, false, bL,
                                                (short)0, acc, false, false);
  }

  const float bias = (c == 0) ? bs[colBase + nc] : 0.f;

  // C/D layout: lane N = lane&15, VGPR v holds M = v + 8*(lane>>4).
  // Row-0 address for this lane's column; rows step by ROW floats.
  float* dst = out + ((long long)tile * 16 + kh * 8) * ROW
                   + (long long)c * H + colBase + nc;
  if (fullTile) {
    #pragma unroll
    for (int v = 0; v < 8; ++v)
      dst[(long long)v * ROW] = acc[v] + bias;
  } else {
    #pragma unroll
    for (int v = 0; v < 8; ++v) {
      long long node = (long long)tile * 16 + kh * 8 + v;
      if (node < N) dst[(long long)v * ROW] = acc[v] + bias;
    }
  }
}

// ---------------------------------------------------------------------------
extern "C" void kernel_launch(void* const* d_in, const int* in_sizes, int n_in,
                              void* d_out, int out_size, void* d_ws, size_t ws_size,
                              hipStream_t stream) {
  const float*     x      = (const float*)d_in[0];
  const long long* ei     = (const long long*)d_in[1];   // int64 edge_index
  const float*     Wsrel  = (const float*)d_in[2];
  const float*     Wsroot = (const float*)d_in[3];
  const float*     bs     = (const float*)d_in[4];
  const float*     Wvrel  = (const float*)d_in[5];
  const float*     Wvroot = (const float*)d_in[6];
  float*           out    = (float*)d_out;

  const int N = in_sizes[0] / (4 * H);
  const int E = in_sizes[1] / 2;

  // 1) zero accumulator (d_out itself)
  long long n4 = (long long)N * (ROW / 4);
  zero_f4<<<2048, 256, 0, stream>>>((float4*)out, n4);

  // 2) raw-feature scatter-add over edges (2 edges per 256-thread block)
  edge_scatter<<<(E + 1) / 2, 256, 0, stream>>>(x, ei, out, E);

  // 3) fused dual-GEMM combine with f32 WMMA, in-place over d_out
  dim3 grid((N + 15) / 16, 4);
  gemm_combine<<<grid, 256, 0, stream>>>(x, Wsrel, Wsroot, bs, Wvrel, Wvroot,
                                         out, N);
}